// SinkhornDistance_65498251264619
// MI455X (gfx1250) — compile-verified
//
#include <hip/hip_runtime.h>
#include <hip/hip_bf16.h>

#define P 2048
#define BATCH 8
#define VCHUNKS 16                  // v-pass row chunks of 128
#define LOG_P 7.624618986159398f    // ln(2048)

typedef float v2f __attribute__((ext_vector_type(2)));
typedef float v8f __attribute__((ext_vector_type(8)));

#if __has_builtin(__builtin_amdgcn_global_load_async_to_lds_b128) && \
    __has_builtin(__builtin_amdgcn_s_wait_asynccnt)
#define SK_ASYNC_LDS 1
typedef int v4i __attribute__((ext_vector_type(4)));
typedef __attribute__((address_space(1))) v4i g1v4;   // global int4
typedef __attribute__((address_space(3))) v4i l3v4;   // LDS int4
#endif

// online log-sum-exp helpers (branchless)
__device__ __forceinline__ void lse_add(float& m, float& s, float val) {
  float nm = fmaxf(m, val);
  s = s * __expf(m - nm) + __expf(val - nm);
  m = nm;
}
__device__ __forceinline__ void lse_merge(float& m, float& s, float om, float os) {
  float nm = fmaxf(m, om);
  s = s * __expf(m - nm) + os * __expf(om - nm);
  m = nm;
}

// ---------------------------------------------------------------------------
// Build D[b][i][j] = |x_i - y_j| via WMMA rank-3 (K=4, padded) update.
// One wave per 16x16 tile; 8 waves (256 thr) per block.
// ---------------------------------------------------------------------------
__global__ void sk_build_D(const float* __restrict__ x, const float* __restrict__ y,
                           float* __restrict__ D) {
  int wave = threadIdx.x >> 5;
  int lane = threadIdx.x & 31;
  int tile = blockIdx.x * 8 + wave;          // 8*128*128 tiles total
  int b  = tile >> 14;
  int rt = (tile >> 7) & 127;
  int ct = tile & 127;
  int kh = lane >> 4;                        // K-half: 0 -> K=0,1 ; 1 -> K=2,3(pad)
  int mn = lane & 15;
  int i0 = rt * 16, j0 = ct * 16;
  const float* xb = x + (size_t)b * P * 3;
  const float* yb = y + (size_t)b * P * 3;

  const float* xp = xb + (size_t)(i0 + mn) * 3;   // A row M=mn, K=2kh,2kh+1
  v2f a;
  a.x = xp[kh * 2];
  a.y = kh ? 0.0f : xp[1];
  const float* yp = yb + (size_t)(j0 + mn) * 3;   // B col N=mn
  v2f bmat;
  bmat.x = yp[kh * 2];
  bmat.y = kh ? 0.0f : yp[1];

  v8f c = {};
  c = __builtin_amdgcn_wmma_f32_16x16x4_f32(false, a, false, bmat,
                                            (short)0, c, false, false);

  float ysq = yp[0]*yp[0] + yp[1]*yp[1] + yp[2]*yp[2];
  size_t bbase = (size_t)b * P * P;
  int j = j0 + mn;
#pragma unroll
  for (int r = 0; r < 8; ++r) {
    int i = i0 + r + 8 * kh;                 // C/D layout: M = r + 8*(lane/16)
    const float* xq = xb + (size_t)i * 3;
    float xsq = xq[0]*xq[0] + xq[1]*xq[1] + xq[2]*xq[2];
    float d2  = xsq + ysq - 2.0f * c[r];
    D[bbase + (size_t)i * P + j] = sqrtf(fmaxf(d2, 0.0f));
  }
}

// ---------------------------------------------------------------------------
// u_new[b,i] = -LOG_P - logsumexp_j( v[b,j] - D[b,i,j] ).
// One wave per row; v staged in LDS via async-load; b128 row streaming with
// 4 independent LSE chains per lane.
// ---------------------------------------------------------------------------
__global__ void sk_u_update(const float* __restrict__ D, float* __restrict__ u,
                            const float* __restrict__ v, unsigned* __restrict__ erru,
                            const int* __restrict__ flag) {
  __shared__ __align__(16) float vbuf[P];
  if (*flag) return;
  int wave = threadIdx.x >> 5, lane = threadIdx.x & 31;
  int gid0 = blockIdx.x * 8;                 // 8 rows per block, same batch (256 blk/batch)
  int b = gid0 >> 11;
  const float* vb = v + b * P;

#ifdef SK_ASYNC_LDS
  {
    int t = threadIdx.x * 4;                 // 256 threads * 2 * b128 = 2048 floats
    __builtin_amdgcn_global_load_async_to_lds_b128(
        (g1v4*)(vb + t), (l3v4*)(vbuf + t), 0, 0);
    __builtin_amdgcn_global_load_async_to_lds_b128(
        (g1v4*)(vb + 1024 + t), (l3v4*)(vbuf + 1024 + t), 0, 0);
    __builtin_amdgcn_s_wait_asynccnt(0);
  }
#else
  for (int t = threadIdx.x; t < P; t += 256) vbuf[t] = vb[t];
#endif
  __syncthreads();

  int gid = gid0 + wave;
  const float4* row = (const float4*)(D + (size_t)gid * P);
  const float4* vv  = (const float4*)vbuf;

  float4 r0 = row[lane];
  float4 w0 = vv[lane];
  float m0 = w0.x - r0.x, m1 = w0.y - r0.y, m2 = w0.z - r0.z, m3 = w0.w - r0.w;
  float s0 = 1.f, s1 = 1.f, s2 = 1.f, s3 = 1.f;
  for (int q = lane + 32; q < P / 4; q += 32) {   // 15 more b128 steps
    float4 r = row[q];
    float4 w = vv[q];
    lse_add(m0, s0, w.x - r.x);
    lse_add(m1, s1, w.y - r.y);
    lse_add(m2, s2, w.z - r.z);
    lse_add(m3, s3, w.w - r.w);
  }
  lse_merge(m0, s0, m1, s1);
  lse_merge(m2, s2, m3, s3);
  lse_merge(m0, s0, m2, s2);
#pragma unroll
  for (int off = 16; off > 0; off >>= 1) {
    float om = __shfl_xor(m0, off, 32);
    float os = __shfl_xor(s0, off, 32);
    lse_merge(m0, s0, om, os);
  }
  float unew = -LOG_P - (m0 + __logf(s0));
  if (lane == 0) {
    float old = u[gid];
    atomicMax(erru + b, __float_as_uint(fabsf(old - unew)));
    u[gid] = unew;
  }
}

// ---------------------------------------------------------------------------
// v-pass stage 1: partial online-LSE over 128-row chunks; each thread owns 4
// columns (b128 reads, 4KB coalesced per row-step per block).
// grid = (colblk=2, chunk=16, batch=8), block = 256.
// ---------------------------------------------------------------------------
__global__ void sk_v_partial(const float* __restrict__ D, const float* __restrict__ u,
                             float* __restrict__ pm, float* __restrict__ ps,
                             const int* __restrict__ flag) {
  if (*flag) return;
  int j4 = blockIdx.x * 256 + threadIdx.x;   // float4-column index: 0..511
  int chunk = blockIdx.y, b = blockIdx.z;
  int i0 = chunk * (P / VCHUNKS);            // 128 rows per chunk
  const float* ub = u + b * P;
  const float4* Db = (const float4*)(D + (size_t)b * P * P);

  float uu = ub[i0];
  float4 r = Db[(size_t)i0 * (P / 4) + j4];
  float m0 = uu - r.x, m1 = uu - r.y, m2 = uu - r.z, m3 = uu - r.w;
  float s0 = 1.f, s1 = 1.f, s2 = 1.f, s3 = 1.f;
  for (int t = 1; t < P / VCHUNKS; ++t) {
    int i = i0 + t;
    float uv = ub[i];                        // uniform -> broadcast load
    float4 rr = Db[(size_t)i * (P / 4) + j4];
    lse_add(m0, s0, uv - rr.x);
    lse_add(m1, s1, uv - rr.y);
    lse_add(m2, s2, uv - rr.z);
    lse_add(m3, s3, uv - rr.w);
  }
  int o = (b * VCHUNKS + chunk) * P + j4 * 4;
  float4 mq = make_float4(m0, m1, m2, m3);
  float4 sq = make_float4(s0, s1, s2, s3);
  *(float4*)(pm + o) = mq;
  *(float4*)(ps + o) = sq;
}

// v-pass stage 2: combine 16 partials per column, write v_new + err.
__global__ void sk_v_combine(const float* __restrict__ pm, const float* __restrict__ ps,
                             float* __restrict__ v, unsigned* __restrict__ errv,
                             const int* __restrict__ flag) {
  if (*flag) return;
  int gid = blockIdx.x * 256 + threadIdx.x;    // 16384
  int b = gid >> 11, j = gid & (P - 1);
  float m = pm[(b * VCHUNKS) * P + j];
  float s = ps[(b * VCHUNKS) * P + j];
#pragma unroll
  for (int c = 1; c < VCHUNKS; ++c) {
    lse_merge(m, s, pm[(b * VCHUNKS + c) * P + j], ps[(b * VCHUNKS + c) * P + j]);
  }
  float vnew = -LOG_P - (m + __logf(s));
  float old = v[gid];
  atomicMax(errv + b, __float_as_uint(fabsf(old - vnew)));
  v[gid] = vnew;
}

// Convergence check: mean-over-batch of per-batch max diffs; set skip flag.
__global__ void sk_conv_check(unsigned* erru, unsigned* errv, int* flag) {
  if (threadIdx.x == 0 && !*flag) {
    float su = 0.f, sv = 0.f;
    for (int b = 0; b < BATCH; ++b) {
      su += __uint_as_float(erru[b]);
      sv += __uint_as_float(errv[b]);
    }
    if (su * 0.125f < 1e-9f && sv * 0.125f < 1e-9f) *flag = 1;
  }
  if (threadIdx.x < BATCH) { erru[threadIdx.x] = 0u; errv[threadIdx.x] = 0u; }
}

// pi = exp(D + u + v) (faithful +D per source); per-block partial of sum(pi*D).
__global__ void sk_pi_partial(const float* __restrict__ D, const float* __restrict__ u,
                              const float* __restrict__ v, float* __restrict__ pi,
                              float* __restrict__ part) {
  size_t base4 = (size_t)blockIdx.x * 1024;    // float4 units; 4096 floats per block
  int b = blockIdx.x >> 10;                    // 1024 blocks per batch
  const float4* D4 = (const float4*)D;
  float4* pi4 = (float4*)pi;
  float acc = 0.f;
#pragma unroll
  for (int e = 0; e < 4; ++e) {
    size_t idx4 = base4 + (size_t)e * 256 + threadIdx.x;
    size_t idx  = idx4 * 4;
    int i = (int)((idx >> 11) & (P - 1));
    int j = (int)(idx & (P - 1));              // j..j+3 in same row
    float uu = u[b * P + i];
    float4 d = D4[idx4];
    float4 p;
    p.x = __expf(d.x + uu + v[b * P + j + 0]);
    p.y = __expf(d.y + uu + v[b * P + j + 1]);
    p.z = __expf(d.z + uu + v[b * P + j + 2]);
    p.w = __expf(d.w + uu + v[b * P + j + 3]);
    pi4[idx4] = p;
    acc += p.x * d.x + p.y * d.y + p.z * d.z + p.w * d.w;
  }
  __shared__ float red[256];
  red[threadIdx.x] = acc;
  __syncthreads();
  for (int st = 128; st > 0; st >>= 1) {
    if (threadIdx.x < st) red[threadIdx.x] += red[threadIdx.x + st];
    __syncthreads();
  }
  if (threadIdx.x == 0) part[blockIdx.x] = red[0];
}

// Deterministic final reduction of 1024 partials per batch.
__global__ void sk_cost_final(const float* __restrict__ part, float* __restrict__ cost) {
  int b = blockIdx.x;
  __shared__ float red[256];
  float acc = 0.f;
  for (int k = threadIdx.x; k < 1024; k += 256) acc += part[b * 1024 + k];
  red[threadIdx.x] = acc;
  __syncthreads();
  for (int st = 128; st > 0; st >>= 1) {
    if (threadIdx.x < st) red[threadIdx.x] += red[threadIdx.x + st];
    __syncthreads();
  }
  if (threadIdx.x == 0) cost[b] = red[0];
}

__global__ void sk_init(float* u, float* v, unsigned* erru, unsigned* errv, int* flag) {
  int gid = blockIdx.x * 256 + threadIdx.x;
  if (gid < BATCH * P) { u[gid] = 0.0f; v[gid] = -LOG_P; }  // v0 = EPS*log_wy
  if (gid < BATCH) { erru[gid] = 0u; errv[gid] = 0u; }
  if (gid == 0) *flag = 0;
}

extern "C" void kernel_launch(void* const* d_in, const int* in_sizes, int n_in,
                              void* d_out, int out_size, void* d_ws, size_t ws_size,
                              hipStream_t stream) {
  const float* x = (const float*)d_in[0];
  const float* y = (const float*)d_in[1];
  float* out  = (float*)d_out;
  float* cost = out;                                  // [8]
  float* pi   = out + 8;                              // [8,2048,2048]
  float* D    = out + 8 + (size_t)BATCH * P * P;      // [8,2048,2048]

  float* w    = (float*)d_ws;
  float* u    = w;                                    // 16384
  float* v    = w + 16384;                            // 16384
  float* pm   = w + 32768;                            // 16*2048*8 = 262144
  float* ps   = w + 294912;                           // 262144
  float* part = w + 557056;                           // 8192
  unsigned* erru = (unsigned*)(w + 565248);           // 8
  unsigned* errv = erru + 8;                          // 8
  int* flag      = (int*)(errv + 8);                  // 1

  sk_init<<<64, 256, 0, stream>>>(u, v, erru, errv, flag);
  sk_build_D<<<16384, 256, 0, stream>>>(x, y, D);
  for (int it = 0; it < 100; ++it) {
    sk_u_update<<<2048, 256, 0, stream>>>(D, u, v, erru, flag);
    sk_v_partial<<<dim3(2, VCHUNKS, 8), 256, 0, stream>>>(D, u, pm, ps, flag);
    sk_v_combine<<<64, 256, 0, stream>>>(pm, ps, v, errv, flag);
    sk_conv_check<<<1, 32, 0, stream>>>(erru, errv, flag);
  }
  sk_pi_partial<<<8192, 256, 0, stream>>>(D, u, v, pi, part);
  sk_cost_final<<<8, 256, 0, stream>>>(part, cost);
}